// SDNE_82635170775050
// MI455X (gfx1250) — compile-verified
//
#include <hip/hip_runtime.h>

// ---------------------------------------------------------------------------
// SDNE-style GraphConv stack for gfx1250 (MI455X, wave32, WMMA).
//   per layer: agg = scatter_add(x[src] -> dst);  h = agg@Wr^T + x@Ws^T + b; leaky_relu
// Bottleneck: edge scatter (~4.5 GB traffic @ 23.3 TB/s HBM / L2-resident
// atomics). GEMMs (~20 GFLOP) run on V_WMMA_F32_16X16X4_F32 (fp32-exact vs
// reference). Block tile = 128x64 so each A fragment feeds 4 WMMAs per path.
// ---------------------------------------------------------------------------

typedef float v2f __attribute__((ext_vector_type(2)));
typedef float v8f __attribute__((ext_vector_type(8)));

#define NNODES    50000
#define NEDGES    800000
#define NEG_SLOPE 0.01f

#define TILE_M     128
#define TILE_N     64   // 4 x 16 WMMA subtiles per wave
#define KT         32
#define LDS_STRIDE 36   // KT + 4 pad: 16B-aligned b128 LDS stores,
                        // row*36 mod 64 spreads banks for column reads.

// ---------------- scatter-add aggregation (segment_sum) --------------------
// One thread per (edge, float4 chunk): consecutive lanes read consecutive
// 16B chunks of the same src row (coalesced gather) and issue f32 atomics to
// consecutive addresses of one dst row (conflict-free in the L2 atomic units).
__global__ __launch_bounds__(256) void scatter_add_kernel(
    const float* __restrict__ x,
    const long long* __restrict__ ei,   // [2, E] int64: row0=src, row1=dst
    float* __restrict__ agg,
    int nedges, int d, int lg_chunks) { // chunks = d/4 = 1 << lg_chunks
  const unsigned t = blockIdx.x * blockDim.x + threadIdx.x;
  const unsigned total = (unsigned)nedges << lg_chunks;
  if (t >= total) return;
  const unsigned e = t >> lg_chunks;
  const unsigned c = t & ((1u << lg_chunks) - 1u);
  const long long s  = ei[e];
  const long long dn = ei[(unsigned)nedges + e];
  const float4 v =
      *reinterpret_cast<const float4*>(x + (size_t)s * d + (size_t)c * 4);
  float* p = agg + (size_t)dn * d + (size_t)c * 4;
  atomicAdd(p + 0, v.x);   // global_atomic_add_f32, no-return
  atomicAdd(p + 1, v.y);
  atomicAdd(p + 2, v.z);
  atomicAdd(p + 3, v.w);
}

// -------- fused dual GEMM + bias + leaky relu (out = agg@Wr^T + x@Ws^T + b)
// grid = (ceil(N/128), d_out/64); block = 256 threads = 8 waves.
// Each wave owns a 16x64 slice -> 4 f32 WMMA accumulators; K tiled by 32
// through LDS. A fragments (agg,x) are reused across the 4 N-subtiles.
__global__ __launch_bounds__(256) void fused_graphconv_gemm(
    const float* __restrict__ agg,
    const float* __restrict__ x,
    const float* __restrict__ Wrel,    // [d_out, d_in] row-major
    const float* __restrict__ Wroot,   // [d_out, d_in] row-major
    const float* __restrict__ bias,    // [d_out]
    float* __restrict__ out,           // [nrows, d_out]
    float* __restrict__ out2,          // optional duplicate (emb), may be null
    int nrows, int d_in, int d_out) {
  __shared__ float ldsAgg[TILE_M * LDS_STRIDE];
  __shared__ float ldsX[TILE_M * LDS_STRIDE];
  __shared__ float ldsWr[TILE_N * LDS_STRIDE];
  __shared__ float ldsWs[TILE_N * LDS_STRIDE];

  const int tid  = threadIdx.x;
  const int wave = tid >> 5;        // 0..7 -> 16-row slice within the block
  const int lane = tid & 31;
  const int half = lane >> 4;       // 0: lanes 0-15, 1: lanes 16-31
  const int l16  = lane & 15;

  const int m0 = blockIdx.x * TILE_M;  // block row base
  const int n0 = blockIdx.y * TILE_N;  // block col base

  v8f acc[4];
#pragma unroll
  for (int nt = 0; nt < 4; ++nt) acc[nt] = v8f{0.f,0.f,0.f,0.f,0.f,0.f,0.f,0.f};

  for (int k0 = 0; k0 < d_in; k0 += KT) {
    __syncthreads();  // protect LDS tiles consumed in the previous K step

    // --- cooperative loads: A tiles 128x32, coalesced float4, 4 passes ---
    {
      const int r0 = tid >> 3;          // 0..31
      const int kc = (tid & 7) * 4;     // 0,4,...,28
      for (int rr = r0; rr < TILE_M; rr += 32) {
        const int grow = m0 + rr;
        float4 va = make_float4(0.f, 0.f, 0.f, 0.f);
        float4 vx = make_float4(0.f, 0.f, 0.f, 0.f);
        if (grow < nrows) {
          va = *reinterpret_cast<const float4*>(
              agg + (size_t)grow * d_in + k0 + kc);
          vx = *reinterpret_cast<const float4*>(
              x + (size_t)grow * d_in + k0 + kc);
        }
        *reinterpret_cast<float4*>(&ldsAgg[rr * LDS_STRIDE + kc]) = va;
        *reinterpret_cast<float4*>(&ldsX[rr * LDS_STRIDE + kc]) = vx;
      }
    }
    // --- weight tiles: 64 output rows x 32 K, both matrices, 2 passes ---
    {
      const int n0l = tid >> 3;         // 0..31
      const int kc  = (tid & 7) * 4;
      for (int n = n0l; n < TILE_N; n += 32) {
        const size_t goff = (size_t)(n0 + n) * d_in + k0 + kc;
        *reinterpret_cast<float4*>(&ldsWr[n * LDS_STRIDE + kc]) =
            *reinterpret_cast<const float4*>(Wrel + goff);
        *reinterpret_cast<float4*>(&ldsWs[n * LDS_STRIDE + kc]) =
            *reinterpret_cast<const float4*>(Wroot + goff);
      }
    }
    __syncthreads();

    // --- fragment pointers ---
    // A (16x4 f32): lane holds M=l16; VGPR j holds K = kf + j, kf = kk + 2*half
    // B (4x16):     lane holds N=l16; same K striping -> read W[n][k] columns.
    const float* aAggB = &ldsAgg[(wave * 16 + l16) * LDS_STRIDE];
    const float* aXB   = &ldsX[(wave * 16 + l16) * LDS_STRIDE];

#pragma unroll
    for (int kk = 0; kk < KT; kk += 4) {
      const int kf = kk + half * 2;
      v2f aA, aX;
      aA.x = aAggB[kf]; aA.y = aAggB[kf + 1];
      aX.x = aXB[kf];   aX.y = aXB[kf + 1];
#pragma unroll
      for (int nt = 0; nt < 4; ++nt) {
        const float* bRelB  = &ldsWr[(nt * 16 + l16) * LDS_STRIDE];
        const float* bRootB = &ldsWs[(nt * 16 + l16) * LDS_STRIDE];
        v2f bR, bS;
        bR.x = bRelB[kf];  bR.y = bRelB[kf + 1];
        bS.x = bRootB[kf]; bS.y = bRootB[kf + 1];
        // D = A*B + C in fp32 (matches reference precision exactly)
        acc[nt] = __builtin_amdgcn_wmma_f32_16x16x4_f32(
            false, aA, false, bR, (short)0, acc[nt], false, false);
        acc[nt] = __builtin_amdgcn_wmma_f32_16x16x4_f32(
            false, aX, false, bS, (short)0, acc[nt], false, false);
      }
    }
  }

  // --- epilogue: bias + leaky relu + store (C layout: M = r + 8*half) ---
  const int rowbase = m0 + wave * 16 + half * 8;
#pragma unroll
  for (int nt = 0; nt < 4; ++nt) {
    const int col = n0 + nt * 16 + l16;
    const float bv = bias[col];
#pragma unroll
    for (int r = 0; r < 8; ++r) {
      const int row = rowbase + r;
      if (row < nrows) {
        float v = acc[nt][r] + bv;
        v = (v > 0.f) ? v : v * NEG_SLOPE;
        out[(size_t)row * d_out + col] = v;
        if (out2) out2[(size_t)row * d_out + col] = v;
      }
    }
  }
}

// ---------------------------------------------------------------------------
extern "C" void kernel_launch(void* const* d_in, const int* in_sizes, int n_in,
                              void* d_out, int out_size, void* d_ws,
                              size_t ws_size, hipStream_t stream) {
  (void)in_sizes; (void)n_in; (void)out_size; (void)ws_size;

  const float* x = (const float*)d_in[0];
  const long long* ei = (const long long*)d_in[1];  // int64 edge_index [2,E]
  float* out = (float*)d_out;

  // workspace: two ping-pong feature buffers + agg buffer (max dim 256)
  float* buf0 = (float*)d_ws;
  float* buf1 = buf0 + (size_t)NNODES * 256;
  float* agg  = buf1 + (size_t)NNODES * 256;

  float* emb_region = out + (size_t)NNODES * 128;  // (h, emb) concatenated

  auto run_layer = [&](const float* hin, int din, int dout, int pi,
                       float* hout, float* hout2) {
    const float* Wr = (const float*)d_in[2 + 3 * pi];
    const float* Ws = (const float*)d_in[3 + 3 * pi];
    const float* b  = (const float*)d_in[4 + 3 * pi];

    hipMemsetAsync(agg, 0, (size_t)NNODES * din * sizeof(float), stream);

    const int lg_chunks = __builtin_ctz(din >> 2);  // chunks = din/4 (pow2)
    const unsigned total = (unsigned)NEDGES << lg_chunks;
    scatter_add_kernel<<<(total + 255u) / 256u, 256, 0, stream>>>(
        hin, ei, agg, NEDGES, din, lg_chunks);

    dim3 grid((NNODES + TILE_M - 1) / TILE_M, dout / TILE_N);
    fused_graphconv_gemm<<<grid, 256, 0, stream>>>(
        agg, hin, Wr, Ws, b, hout, hout2, NNODES, din, dout);
  };

  run_layer(x,    128, 256, 0, buf0, nullptr);      // encoder 1
  run_layer(buf0, 256,  64, 1, buf1, emb_region);   // encoder 2 -> emb
  run_layer(buf1,  64, 256, 2, buf0, nullptr);      // decoder 1
  run_layer(buf0, 256, 128, 3, out,  nullptr);      // decoder 2 -> h
}